// SingleLevelConditionalDriftingLoss_88218628260543
// MI455X (gfx1250) — compile-verified
//
#include <hip/hip_runtime.h>
#include <math.h>

// Problem constants (from reference setup_inputs)
#define BB 4
#define NN 8
#define MM 8
#define CC 3
#define HH 128
#define WW 128
#define CG 128          // encoder out channels
#define HS 16           // conv out height
#define WSP 16          // conv out width
#define LL 256          // HS*WSP
#define KK 192          // C*8*8 patch size
#define NROWS (BB*NN*LL)   // 8192 rows in g_bank / p_bank
#define TAU_F 0.1f

typedef __attribute__((ext_vector_type(2))) float v2f;
typedef __attribute__((ext_vector_type(8))) float v8f;

// ---------------------------------------------------------------------------
// Encode: stride-8 8x8 VALID conv == per-(img,l) dot of a 192-elem patch with
// each of 128 filter rows. One block per (img, l); thread = output channel.
// Optionally subtracts the x_up encoding (conv is linear).
// Output layout: out[img][l][cg], cg contiguous (row = 128-dim feature).
// ---------------------------------------------------------------------------
__global__ void __launch_bounds__(128)
encode_kernel(const float* __restrict__ x, const float* __restrict__ Wenc,
              const float* __restrict__ sub, float* __restrict__ out,
              int imgsPerB) {
  __shared__ float patch[KK];
  const int blk = blockIdx.x;          // img*LL + l
  const int img = blk / LL;
  const int l   = blk % LL;
  const int hs = l / WSP, ws = l % WSP;
  const float* xi = x + (size_t)img * (CC * HH * WW);
  for (int k = threadIdx.x; k < KK; k += 128) {
    int c  = k >> 6;            // /64
    int kh = (k >> 3) & 7;
    int kw = k & 7;
    patch[k] = xi[c * (HH * WW) + (hs * 8 + kh) * WW + (ws * 8 + kw)];
  }
  __syncthreads();
  const int cg = threadIdx.x;
  const float* wrow = Wenc + (size_t)cg * KK;   // W_enc[cg][c][kh][kw] flat
  float acc = 0.f;
  #pragma unroll 8
  for (int k = 0; k < KK; ++k) acc = fmaf(patch[k], wrow[k], acc);
  if (sub) {
    int b = img / imgsPerB;
    acc -= sub[((size_t)b * LL + l) * CG + cg];
  }
  out[((size_t)img * LL + l) * CG + cg] = acc;
}

// ---------------------------------------------------------------------------
// Squared L2 norm of each 128-dim row. One wave per row.
// ---------------------------------------------------------------------------
__global__ void __launch_bounds__(256)
rownorm_kernel(const float* __restrict__ A, float* __restrict__ nrm, int nrows) {
  const int row  = blockIdx.x * (blockDim.x >> 5) + (threadIdx.x >> 5);
  const int lane = threadIdx.x & 31;
  if (row >= nrows) return;
  const float* r = A + (size_t)row * CG;
  float s = 0.f;
  for (int c = lane; c < CG; c += 32) s = fmaf(r[c], r[c], s);
  for (int off = 16; off; off >>= 1) s += __shfl_xor(s, off, 32);
  if (lane == 0) nrm[row] = s;
}

// ---------------------------------------------------------------------------
// The big one: pdist(g_bank, p_bank).sum() without materializing 8192x8192.
// Dot-products via V_WMMA_F32_16X16X4_F32 (fp32 matrix pipe, reference
// precision), fused d = sqrt(max(|g|^2+|p|^2-2*dot, 1e-12)) and double-
// precision block sums. One wave computes a 64x64 macro-tile: 16 WMMA per
// 8 b64 loads per K-step (4x the compute density of a 32x32 tile, 1/4 the
// redundant cache traffic). 16 v8f accumulators = 128 VGPRs, no spill.
// Grid: 128x128 tiles of 64x64 = 16384 waves = 2048 blocks x 8 waves.
// ---------------------------------------------------------------------------
__global__ void __launch_bounds__(256)
pdist_sum_kernel(const float* __restrict__ G, const float* __restrict__ P,
                 const float* __restrict__ gn, const float* __restrict__ pn,
                 double* __restrict__ partial) {
  __shared__ double red[256];
  const int lane = threadIdx.x & 31;
  const int wave = threadIdx.x >> 5;
  const int tile = blockIdx.x * 8 + wave;   // [0, 16384)
  const int r0 = (tile >> 7) * 64;          // row block in G
  const int c0 = (tile & 127) * 64;         // row block in P
  const int lr    = lane & 15;
  const int khalf = (lane >> 4) << 1;       // lanes 16-31 hold K+2,K+3

  v8f acc[4][4];
  #pragma unroll
  for (int ti = 0; ti < 4; ++ti)
    #pragma unroll
    for (int tj = 0; tj < 4; ++tj) acc[ti][tj] = (v8f){};

  // Base pointers; the 16-row group stride (16*CG*4 = 8192 B) folds into the
  // 24-bit instruction offset, so only two base addresses are live.
  const float* Gp = G + (size_t)(r0 + lr) * CG + khalf;
  const float* Pp = P + (size_t)(c0 + lr) * CG + khalf;

  #pragma unroll 2
  for (int k = 0; k < CG; k += 4) {
    v2f a[4], b[4];
    #pragma unroll
    for (int t = 0; t < 4; ++t) {
      a[t] = *(const v2f*)(Gp + (size_t)t * 16 * CG + k);
      b[t] = *(const v2f*)(Pp + (size_t)t * 16 * CG + k);
    }
    #pragma unroll
    for (int ti = 0; ti < 4; ++ti)
      #pragma unroll
      for (int tj = 0; tj < 4; ++tj)
        acc[ti][tj] = __builtin_amdgcn_wmma_f32_16x16x4_f32(
            false, a[ti], false, b[tj], (short)0, acc[ti][tj], false, false);
  }

  // D layout: vgpr i, lane<16 -> (m=i, n=lane); lane>=16 -> (m=i+8, n=lane-16)
  const int n_lo  = lane & 15;
  const int m_add = (lane >> 4) << 3;
  double s = 0.0;
  #pragma unroll
  for (int ti = 0; ti < 4; ++ti) {
    #pragma unroll
    for (int tj = 0; tj < 4; ++tj) {
      const float pnv = pn[c0 + 16 * tj + n_lo];
      #pragma unroll
      for (int i = 0; i < 8; ++i) {
        const float gnv = gn[r0 + 16 * ti + m_add + i];
        s += (double)sqrtf(fmaxf(gnv + pnv - 2.0f * acc[ti][tj][i], 1e-12f));
      }
    }
  }
  red[threadIdx.x] = s;
  __syncthreads();
  for (int st = 128; st; st >>= 1) {
    if (threadIdx.x < st) red[threadIdx.x] += red[threadIdx.x + st];
    __syncthreads();
  }
  if (threadIdx.x == 0) partial[blockIdx.x] = red[0];
}

// ---------------------------------------------------------------------------
// Reduce pdist partials -> feat_scale; also resolve lambda(epoch).
// scal[0] = feat_scale, scal[1] = lambda_same_neg
// ---------------------------------------------------------------------------
__global__ void __launch_bounds__(256)
feat_scale_kernel(const double* __restrict__ partial, int n,
                  const int* __restrict__ epoch, float* __restrict__ scal) {
  __shared__ double red[256];
  double s = 0.0;
  for (int i = threadIdx.x; i < n; i += 256) s += partial[i];
  red[threadIdx.x] = s;
  __syncthreads();
  for (int st = 128; st; st >>= 1) {
    if (threadIdx.x < st) red[threadIdx.x] += red[threadIdx.x + st];
    __syncthreads();
  }
  if (threadIdx.x == 0) {
    const double mean = red[0] / ((double)NROWS * (double)NROWS);
    float fs = (float)(mean / 11.313708498984761);   // / sqrt(128)
    fs = fmaxf(fmaxf(fs, 1e-6f), 1e-4f);
    scal[0] = fs;
    const int e = epoch[0];
    float lam;
    if (e <= 5)       lam = 0.0f;
    else if (e <= 15) lam = ((float)(e - 5) / 10.0f) * 0.5f;
    else              lam = 0.5f;
    scal[1] = lam;
  }
}

// ---------------------------------------------------------------------------
// Conditional drift + sum(v^2). Thread per (b, l, n): two passes over the
// 128 channels (distances -> softmax weights -> v on the fly, never stored).
// 8192 threads total; double-precision block partials.
// ---------------------------------------------------------------------------
__global__ void __launch_bounds__(256)
drift_kernel(const float* __restrict__ G, const float* __restrict__ P,
             const float* __restrict__ scal, double* __restrict__ partial) {
  __shared__ double red[256];
  const int tid = blockIdx.x * 256 + threadIdx.x;   // [0, 8192)
  const int b   = tid / (LL * NN);
  const int rem = tid % (LL * NN);
  const int l   = rem / NN;
  const int n   = rem % NN;
  const float inv_fs = 1.0f / scal[0];
  const float lam    = scal[1];

  const float* xrow[NN];
  const float* yrow[MM];
  #pragma unroll
  for (int m = 0; m < NN; ++m) xrow[m] = G + ((size_t)(b * NN + m) * LL + l) * CG;
  #pragma unroll
  for (int m = 0; m < MM; ++m) yrow[m] = P + ((size_t)(b * MM + m) * LL + l) * CG;
  const float* xn = G + ((size_t)(b * NN + n) * LL + l) * CG;

  float d2xy[MM], d2xx[NN];
  #pragma unroll
  for (int m = 0; m < MM; ++m) { d2xy[m] = 0.f; d2xx[m] = 0.f; }

  for (int c = 0; c < CG; ++c) {
    const float xc = xn[c] * inv_fs;
    #pragma unroll
    for (int m = 0; m < MM; ++m) {
      const float dy = xc - yrow[m][c] * inv_fs;
      d2xy[m] = fmaf(dy, dy, d2xy[m]);
      const float dx = xc - xrow[m][c] * inv_fs;
      d2xx[m] = fmaf(dx, dx, d2xx[m]);
    }
  }

  // softmax(-dist/TAU) over positives
  float wpos[MM], wsame[NN];
  float mx = -1e30f;
  #pragma unroll
  for (int m = 0; m < MM; ++m) {
    const float lg = -sqrtf(fmaxf(d2xy[m], 1e-12f)) / TAU_F;
    wpos[m] = lg; mx = fmaxf(mx, lg);
  }
  float se = 0.f;
  #pragma unroll
  for (int m = 0; m < MM; ++m) { wpos[m] = __expf(wpos[m] - mx); se += wpos[m]; }
  float rs = 1.f / se;
  #pragma unroll
  for (int m = 0; m < MM; ++m) wpos[m] *= rs;

  // softmax(-(dist + 1e6*eye)/TAU) over same-set
  mx = -1e30f;
  #pragma unroll
  for (int m = 0; m < NN; ++m) {
    const float d  = sqrtf(fmaxf(d2xx[m], 1e-12f)) + ((m == n) ? 1e6f : 0.f);
    const float lg = -d / TAU_F;
    wsame[m] = lg; mx = fmaxf(mx, lg);
  }
  se = 0.f;
  #pragma unroll
  for (int m = 0; m < NN; ++m) { wsame[m] = __expf(wsame[m] - mx); se += wsame[m]; }
  rs = 1.f / se;
  #pragma unroll
  for (int m = 0; m < NN; ++m) wsame[m] *= rs;

  double acc = 0.0;
  for (int c = 0; c < CG; ++c) {
    const float xc = xn[c] * inv_fs;
    float vp = -xc, vs = -xc;
    #pragma unroll
    for (int m = 0; m < MM; ++m) vp = fmaf(wpos[m], yrow[m][c] * inv_fs, vp);
    #pragma unroll
    for (int m = 0; m < NN; ++m) vs = fmaf(wsame[m], xrow[m][c] * inv_fs, vs);
    const float v = vp - lam * vs;   // LAMBDA_POS = 1
    acc += (double)(v * v);
  }
  red[threadIdx.x] = acc;
  __syncthreads();
  for (int st = 128; st; st >>= 1) {
    if (threadIdx.x < st) red[threadIdx.x] += red[threadIdx.x + st];
    __syncthreads();
  }
  if (threadIdx.x == 0) partial[blockIdx.x] = red[0];
}

// ---------------------------------------------------------------------------
// Final: loss = (DRIFT_STEP/clip(sqrt(mean v^2 + eps),0.1,10))^2 * mean v^2
// ---------------------------------------------------------------------------
__global__ void __launch_bounds__(64)
final_kernel(const double* __restrict__ partial, int n, float* __restrict__ out) {
  __shared__ double red[64];
  double s = 0.0;
  for (int i = threadIdx.x; i < n; i += 64) s += partial[i];
  red[threadIdx.x] = s;
  __syncthreads();
  for (int st = 32; st; st >>= 1) {
    if (threadIdx.x < st) red[threadIdx.x] += red[threadIdx.x + st];
    __syncthreads();
  }
  if (threadIdx.x == 0) {
    const double meanv2 = red[0] / (double)((size_t)BB * LL * NN * CG);
    const float ds = sqrtf((float)meanv2 + 1e-6f);
    const float nf = fminf(fmaxf(ds, 0.1f), 10.0f);
    const float r  = 0.1f / nf;
    out[0] = r * r * (float)meanv2;
  }
}

// ---------------------------------------------------------------------------
// Orchestration. Workspace layout (floats unless noted), ~9.1 MB total:
//   G[8192*128] | P[8192*128] | Up[4*256*128] | gn[8192] | pn[8192] |
//   scal[4] | part1 double[2048] | part2 double[32]
// ---------------------------------------------------------------------------
extern "C" void kernel_launch(void* const* d_in, const int* in_sizes, int n_in,
                              void* d_out, int out_size, void* d_ws, size_t ws_size,
                              hipStream_t stream) {
  const float* x_gen = (const float*)d_in[0];
  const float* x_pos = (const float*)d_in[1];
  const float* x_up  = (const float*)d_in[2];
  const float* Wenc  = (const float*)d_in[3];
  const int*   epoch = (const int*)d_in[4];
  float* out = (float*)d_out;

  float* G  = (float*)d_ws;
  float* P  = G  + (size_t)NROWS * CG;
  float* Up = P  + (size_t)NROWS * CG;
  float* gn = Up + (size_t)BB * LL * CG;
  float* pn = gn + NROWS;
  float* scal = pn + NROWS;            // 4 floats (2 used + pad -> keeps 8B align)
  double* part1 = (double*)(scal + 4); // 2048 doubles
  double* part2 = part1 + 2048;        // 32 doubles

  // 1) Encode x_up (no subtraction), then x_gen/x_pos minus the up encoding.
  encode_kernel<<<BB * LL, 128, 0, stream>>>(x_up, Wenc, nullptr, Up, 1);
  encode_kernel<<<BB * NN * LL, 128, 0, stream>>>(x_gen, Wenc, Up, G, NN);
  encode_kernel<<<BB * MM * LL, 128, 0, stream>>>(x_pos, Wenc, Up, P, MM);

  // 2) Row norms for the distance expansion.
  rownorm_kernel<<<NROWS / 8, 256, 0, stream>>>(G, gn, NROWS);
  rownorm_kernel<<<NROWS / 8, 256, 0, stream>>>(P, pn, NROWS);

  // 3) 8192x8192x128 fp32 WMMA GEMM fused with sqrt-distance + mean reduction.
  pdist_sum_kernel<<<2048, 256, 0, stream>>>(G, P, gn, pn, part1);

  // 4) feat_scale + lambda(epoch) scalars.
  feat_scale_kernel<<<1, 256, 0, stream>>>(part1, 2048, epoch, scal);

  // 5) Conditional drift, sum(v^2).
  drift_kernel<<<32, 256, 0, stream>>>(G, P, scal, part2);

  // 6) Final scalar loss.
  final_kernel<<<1, 64, 0, stream>>>(part2, 32, out);
}